// VoxelRenderLayer_11957188952688
// MI455X (gfx1250) — compile-verified
//
#include <hip/hip_runtime.h>
#include <hip/hip_bf16.h>

typedef __attribute__((ext_vector_type(2))) float v2f;
typedef __attribute__((ext_vector_type(8))) float v8f;

#define VOX3   (128*128*128)
#define NS     192
#define EPSF   1.0e-4f

__device__ __forceinline__ float fetch_vol(const float* __restrict__ vol,
                                           int iz, int iy, int ix) {
    return vol[(iz*128 + iy)*128 + ix];
}

// ---------------------------------------------------------------------------
// Kernel 1: rotated trilinear resample. One wave handles 16 output voxels per
// tile; the trilinear 8-corner weighted sum is evaluated with two chained
// V_WMMA_F32_16X16X4_F32 ops (B = all-ones -> row-sum; only the documented
// A and C/D layouts are relied upon).
//   lane m    (0..15): corners dy=0, K = {dz0dx0, dz0dx1} then {dz1dx0, dz1dx1}
//   lane m+16        : same with dy=1
// ---------------------------------------------------------------------------
__global__ __launch_bounds__(256) void rotate_resample_wmma(
    const float* __restrict__ vox,      // (4,1,128,128,128)
    const float* __restrict__ rot,      // (4,3,4)
    float* __restrict__ out)            // (4,128,128,128) workspace
{
    const int lane = threadIdx.x & 31;
    const int wave = threadIdx.x >> 5;
    const int m    = lane & 15;
    const int half = lane >> 4;                     // dy for this lane

    const long tile = (long)blockIdx.x * 8 + wave;  // 524288 tiles of 16 voxels
    const int  wt   = (int)(tile & 7);              // 8 tiles per w-row
    const long row  = tile >> 3;                    // ((n*128)+d)*128 + h
    const int  h    = (int)(row & 127);
    const int  d    = (int)((row >> 7) & 127);
    const int  n    = (int)(row >> 14);
    const int  w    = wt*16 + m;

    // normalized coords (jnp.linspace(-1,1,128))
    const float sc = 2.0f/127.0f;
    const float X = -1.0f + w*sc;
    const float Y = -1.0f + h*sc;
    const float Z = -1.0f + d*sc;

    const float* th = rot + n*12;                   // theta[n] (3x4)
    const float gx = th[0]*X + th[1]*Y + th[2]*Z  + th[3];
    const float gy = th[4]*X + th[5]*Y + th[6]*Z  + th[7];
    const float gz = th[8]*X + th[9]*Y + th[10]*Z + th[11];

    // to voxel coords: (g+1)*0.5*(W-1)
    const float x = (gx + 1.0f)*63.5f;
    const float y = (gy + 1.0f)*63.5f;
    const float z = (gz + 1.0f)*63.5f;

    const float x0f = floorf(x), y0f = floorf(y), z0f = floorf(z);
    const float fx = x - x0f, fy = y - y0f, fz = z - z0f;
    const int x0 = (int)x0f, y0 = (int)y0f, z0 = (int)z0f;
    const int x1 = x0 + 1,  z1 = z0 + 1;

    // y handled per lane-half (dy = half)
    const int   iy  = y0 + half;
    float wy = half ? fy : (1.0f - fy);
    wy = (iy >= 0 && iy < 128) ? wy : 0.0f;
    const int iyc = min(max(iy, 0), 127);

    // validity folded into weights (matches reference's weight*valid)
    const float wx0 = (x0 >= 0 && x0 < 128) ? (1.0f - fx) : 0.0f;
    const float wx1 = (x1 >= 0 && x1 < 128) ? fx          : 0.0f;
    const float wz0 = (z0 >= 0 && z0 < 128) ? (1.0f - fz) : 0.0f;
    const float wz1 = (z1 >= 0 && z1 < 128) ? fz          : 0.0f;
    const int x0c = min(max(x0, 0), 127), x1c = min(max(x1, 0), 127);
    const int z0c = min(max(z0, 0), 127), z1c = min(max(z1, 0), 127);

    const float* vol = vox + (long)n*VOX3;
    const float v00 = fetch_vol(vol, z0c, iyc, x0c);
    const float v01 = fetch_vol(vol, z0c, iyc, x1c);
    const float v10 = fetch_vol(vol, z1c, iyc, x0c);
    const float v11 = fetch_vol(vol, z1c, iyc, x1c);

    v2f a1; a1.x = wz0*wy*wx0*v00;  a1.y = wz0*wy*wx1*v01;   // K0,K1 (dz=0)
    v2f a2; a2.x = wz1*wy*wx0*v10;  a2.y = wz1*wy*wx1*v11;   // K2,K3 (dz=1)
    v2f bones; bones.x = 1.0f; bones.y = 1.0f;               // all-ones B

    v8f c = {};
    c = __builtin_amdgcn_wmma_f32_16x16x4_f32(false, a1, false, bones,
                                              (short)0, c, false, false);
    c = __builtin_amdgcn_wmma_f32_16x16x4_f32(false, a2, false, bones,
                                              (short)0, c, false, false);

    // D layout: VGPR r, lane l -> M = r + (l>=16 ? 8 : 0); all columns equal.
    const int q = lane & 7;
    float val = c[0];
    val = (q == 1) ? c[1] : val;
    val = (q == 2) ? c[2] : val;
    val = (q == 3) ? c[3] : val;
    val = (q == 4) ? c[4] : val;
    val = (q == 5) ? c[5] : val;
    val = (q == 6) ? c[6] : val;
    val = (q == 7) ? c[7] : val;

    if ((lane & 8) == 0) {                 // lanes 0-7 -> M=0..7, 16-23 -> M=8..15
        const int M = q + (half << 3);
        out[row*128 + wt*16 + M] = val;
    }
}

// ---------------------------------------------------------------------------
// Kernel 2: ray march. One thread per ray (4*128*128 = 65536 rays), sequential
// transmittance scan over 192 samples, trilinear samples from the L2-resident
// rotated volume, depth weights staged in LDS.
// ---------------------------------------------------------------------------
__device__ __forceinline__ float trilinear(const float* __restrict__ vol,
                                           float gx, float gy, float gz)
{
    const float x = (gx + 1.0f)*63.5f;
    const float y = (gy + 1.0f)*63.5f;
    const float z = (gz + 1.0f)*63.5f;
    const float x0f = floorf(x), y0f = floorf(y), z0f = floorf(z);
    const float fx = x - x0f, fy = y - y0f, fz = z - z0f;
    const int x0 = (int)x0f, y0 = (int)y0f, z0 = (int)z0f;
    const int x1 = x0 + 1, y1 = y0 + 1, z1 = z0 + 1;
    const float wx0 = (x0 >= 0 && x0 < 128) ? (1.0f - fx) : 0.0f;
    const float wx1 = (x1 >= 0 && x1 < 128) ? fx          : 0.0f;
    const float wy0 = (y0 >= 0 && y0 < 128) ? (1.0f - fy) : 0.0f;
    const float wy1 = (y1 >= 0 && y1 < 128) ? fy          : 0.0f;
    const float wz0 = (z0 >= 0 && z0 < 128) ? (1.0f - fz) : 0.0f;
    const float wz1 = (z1 >= 0 && z1 < 128) ? fz          : 0.0f;
    const int x0c = min(max(x0,0),127), x1c = min(max(x1,0),127);
    const int y0c = min(max(y0,0),127), y1c = min(max(y1,0),127);
    const int z0c = min(max(z0,0),127), z1c = min(max(z1,0),127);
    float acc;
    acc  = wz0*wy0*wx0 * fetch_vol(vol, z0c, y0c, x0c);
    acc += wz0*wy0*wx1 * fetch_vol(vol, z0c, y0c, x1c);
    acc += wz0*wy1*wx0 * fetch_vol(vol, z0c, y1c, x0c);
    acc += wz0*wy1*wx1 * fetch_vol(vol, z0c, y1c, x1c);
    acc += wz1*wy0*wx0 * fetch_vol(vol, z1c, y0c, x0c);
    acc += wz1*wy0*wx1 * fetch_vol(vol, z1c, y0c, x1c);
    acc += wz1*wy1*wx0 * fetch_vol(vol, z1c, y1c, x0c);
    acc += wz1*wy1*wx1 * fetch_vol(vol, z1c, y1c, x1c);
    return acc;
}

__global__ __launch_bounds__(256) void raymarch(
    const float* __restrict__ vrot,     // (4,128,128,128) workspace
    const float* __restrict__ grid,     // (4,192,128,128,3)
    const float* __restrict__ dw,       // (192,)
    float* __restrict__ out)            // [sil(4,128,128) | depth(4,128,128)]
{
    __shared__ float sdw[NS];
    for (int t = threadIdx.x; t < NS; t += blockDim.x) sdw[t] = dw[t];
    __syncthreads();

    const int r = blockIdx.x*256 + threadIdx.x;     // 65536 rays
    const int j = r & 127;
    const int i = (r >> 7) & 127;
    const int n = r >> 14;
    const float* vol = vrot + (long)n*VOX3;

    // grid flat index for (n, k, i, j, 0); k-stride = 128*128*3
    long g = ((long)n*(NS*128*128) + (long)i*128 + j) * 3;

    float T = 1.0f, sil = 0.0f, dep = 0.0f;
    for (int k = 0; k < NS; ++k, g += (long)128*128*3) {
        const float gx = grid[g + 0];
        const float gy = grid[g + 1];
        const float gz = grid[g + 2];
        const float v = trilinear(vol, gx, gy, gz);
        const float p = fminf(fmaxf(v, EPSF), 1.0f - EPSF);
        const float stop = p * T;
        sil += stop;
        dep += stop * sdw[k];
        T   *= (1.0f - p);
    }
    dep += fminf(fmaxf(T, EPSF), 1.0f - EPSF) * 3.0f;   // bg = clip(q_last)*(DIST+1)

    // outputs are swapaxes(2,3): out[n, j, i]
    const int o = n*16384 + j*128 + i;
    out[o]         = sil;
    out[65536 + o] = dep;
}

extern "C" void kernel_launch(void* const* d_in, const int* in_sizes, int n_in,
                              void* d_out, int out_size, void* d_ws, size_t ws_size,
                              hipStream_t stream)
{
    const float* vox  = (const float*)d_in[0];   // (4,1,128,128,128)
    const float* rot  = (const float*)d_in[1];   // (4,3,4)
    const float* grid = (const float*)d_in[2];   // (4,192,128,128,3)
    const float* dw   = (const float*)d_in[3];   // (192,)
    float* vrot = (float*)d_ws;                  // 4*128^3 floats = 32 MiB scratch
    float* out  = (float*)d_out;

    // 524288 tiles of 16 voxels, 8 waves (tiles) per 256-thread block
    rotate_resample_wmma<<<65536, 256, 0, stream>>>(vox, rot, vrot);
    // 65536 rays, one per thread
    raymarch<<<256, 256, 0, stream>>>(vrot, grid, dw, out);

    (void)in_sizes; (void)n_in; (void)out_size; (void)ws_size;
}